// Search_11922829213965
// MI455X (gfx1250) — compile-verified
//
#include <hip/hip_runtime.h>
#include <hip/hip_bf16.h>
#include <math.h>

// ---------------------------------------------------------------------------
// Cosine-similarity search (1 query x 200000 features, D=1024, top-100).
//
// Roofline: 819 MB of features must stream once -> ~35 us at 23.3 TB/s.
// Pure bandwidth problem; WMMA_F32_16X16X4_F32 does the matvec in the matrix
// pipe while the stream is in flight, row norms fused in the same pass.
// Top-k pass works on an 800 KB sims array (L2-resident).
// ---------------------------------------------------------------------------

typedef __attribute__((ext_vector_type(2))) float v2f;
typedef __attribute__((ext_vector_type(8))) float v8f;

#define DIM   1024
#define EPS   1e-8f

// ---------------------------------------------------------------------------
// Kernel 1: sims[n] = dot(F[n], q) / max(||F[n]|| * ||q||, eps)
//
// Block = 256 threads = 8 waves; each wave owns 16 feature rows (block: 128).
// WMMA f32 16x16x4 mapping per wave:
//   A (16x4): lanes 0-15 supply K-slots 0,1 (VGPR0,1), lanes 16-31 K-slots 2,3.
//   We load per lane a float4 from its row: lanes 0-15 at column c, lanes
//   16-31 at column c+4.  Then K-slots {0,1,2,3} of WMMA #1 hold columns
//   {c, c+1, c+4, c+5} and WMMA #2 holds {c+2, c+3, c+6, c+7}.
//   B columns are all identical (the query), and each lane's B value is the
//   query at exactly the same column its A value came from -> B is just the
//   query float4 loaded with the same lane addressing (from LDS).
//   After the K loop D[m][*] = dot(row m, q): rows 0-7 live in acc[0..7] of
//   lanes 0-15, rows 8-15 in acc[0..7] of lanes 16-31.
// ---------------------------------------------------------------------------
__global__ __launch_bounds__(256) void sim_kernel(const float* __restrict__ query,
                                                  const float* __restrict__ features,
                                                  float* __restrict__ sims,
                                                  int n)
{
    __shared__ __align__(16) float q_lds[DIM];
    __shared__ float qn2_lds;

    const int tid = threadIdx.x;

    // Stage the 4 KB query into LDS (256 threads x float4) + query norm.
    float4 qq = ((const float4*)query)[tid];
    ((float4*)q_lds)[tid] = qq;
    if (tid == 0) qn2_lds = 0.f;
    __syncthreads();
    atomicAdd(&qn2_lds, qq.x * qq.x + qq.y * qq.y + qq.z * qq.z + qq.w * qq.w);
    __syncthreads();
    const float q_norm = sqrtf(qn2_lds);

    const int wave   = tid >> 5;
    const int lane   = tid & 31;
    const int laneLo = lane & 15;
    const int laneHi = lane >> 4;          // 0: K 0..3 side, 1: K 4..7 side
    const int row0   = blockIdx.x * 128 + wave * 16;

    int myRow   = row0 + laneLo;
    int loadRow = (myRow < n) ? myRow : (n - 1);   // clamp tail, discard on write
    const float* rp = features + (size_t)loadRow * DIM + laneHi * 4;

    v8f  acc  = {0.f, 0.f, 0.f, 0.f, 0.f, 0.f, 0.f, 0.f};
    float nrm2 = 0.f;

#pragma unroll 4
    for (int c = 0; c < DIM; c += 8) {
        float4 a  = *(const float4*)(rp + c);                  // global_load_b128
        float4 qv = *(const float4*)(&q_lds[c + laneHi * 4]);  // ds_load_b128

        v2f a01 = {a.x, a.y}, a23 = {a.z, a.w};
        v2f b01 = {qv.x, qv.y}, b23 = {qv.z, qv.w};

        // D = A*B + C, f32 16x16x4 (8-arg form: neg_a, A, neg_b, B, c_mod, C, reuse_a, reuse_b)
        acc = __builtin_amdgcn_wmma_f32_16x16x4_f32(false, a01, false, b01,
                                                    (short)0, acc, false, false);
        acc = __builtin_amdgcn_wmma_f32_16x16x4_f32(false, a23, false, b23,
                                                    (short)0, acc, false, false);

        // Fused row-norm partial (this lane covers half of row laneLo's columns)
        nrm2 += a.x * a.x + a.y * a.y + a.z * a.z + a.w * a.w;
    }

    // Row norm^2: combine the two half-row partials (lanes L and L+16).
    nrm2 += __shfl_xor(nrm2, 16, 32);
    // After this, lane L holds norm2 of row (row0 + (L & 15)).

    // Writers: lanes 0-7 emit rows row0+0..7 (from acc of low half),
    //          lanes 16-23 emit rows row0+8..15 (from acc of high half).
    const int sel = lane & 7;
    float dot = 0.f;
#pragma unroll
    for (int v = 0; v < 8; ++v)
        if (sel == v) dot = acc[v];

    // Writer lane L needs norm2 of its output row:
    //   lanes 0-7  -> row sel     -> held by lane sel   (= L)
    //   lanes 16-23-> row 8+sel   -> held by lane 8+sel (= L-8)
    int nsrc = (lane < 16) ? lane : (lane - 8);
    float rown2 = __shfl(nrm2, nsrc & 31, 32);

    const bool writer = (lane < 8) || (lane >= 16 && lane < 24);
    const int outRow  = row0 + sel + (laneHi ? 8 : 0);
    if (writer && outRow < n) {
        float denom = fmaxf(sqrtf(rown2) * q_norm, EPS);
        sims[outRow] = dot / denom;
    }
}

// ---------------------------------------------------------------------------
// Kernel 2: iterative top-k (descending, ties -> lowest index like lax.top_k).
// One block of 1024 threads; each thread owns a contiguous slice and caches
// its local argmax; per round, wave 0 reduces the 1024 candidates (LDS scan +
// 5 shfl_xor steps) and only the losing slice rescans. sims stays in L2.
// ---------------------------------------------------------------------------
__global__ __launch_bounds__(1024) void topk_kernel(float* __restrict__ sims,
                                                    float* __restrict__ outV,
                                                    int* __restrict__ outI,
                                                    int n, int topk)
{
    __shared__ float sv[1024];
    __shared__ int   si[1024];
    __shared__ int   s_widx;

    const int t  = threadIdx.x;
    const int ch = (n + 1023) >> 10;          // slice length per thread
    const int beg = t * ch;
    const int end = (beg + ch < n) ? (beg + ch) : n;

    float bv = -INFINITY;
    int   bi = 0x7fffffff;
    for (int i = beg; i < end; ++i) {
        float v = sims[i];
        if (v > bv || (v == bv && i < bi)) { bv = v; bi = i; }
    }
    sv[t] = bv;
    si[t] = bi;

    for (int k = 0; k < topk; ++k) {
        __syncthreads();
        if (t < 32) {
            float v  = -INFINITY;
            int  idx = 0x7fffffff;
#pragma unroll
            for (int j = 0; j < 32; ++j) {
                float cv = sv[t + 32 * j];
                int   ci = si[t + 32 * j];
                if (cv > v || (cv == v && ci < idx)) { v = cv; idx = ci; }
            }
#pragma unroll
            for (int off = 16; off > 0; off >>= 1) {
                float ov = __shfl_xor(v, off, 32);
                int   oi = __shfl_xor(idx, off, 32);
                if (ov > v || (ov == v && oi < idx)) { v = ov; idx = oi; }
            }
            if (t == 0) { outV[k] = v; outI[k] = idx; s_widx = idx; }
        }
        __syncthreads();
        const int widx = s_widx;
        if (widx >= beg && widx < end) {       // only the losing slice rescans
            sims[widx] = -INFINITY;
            float nv = -INFINITY;
            int   ni = 0x7fffffff;
            for (int i = beg; i < end; ++i) {
                float v = sims[i];
                if (v > nv || (v == nv && i < ni)) { nv = v; ni = i; }
            }
            sv[t] = nv;
            si[t] = ni;
        }
    }
}

// ---------------------------------------------------------------------------
extern "C" void kernel_launch(void* const* d_in, const int* in_sizes, int n_in,
                              void* d_out, int out_size, void* d_ws, size_t ws_size,
                              hipStream_t stream)
{
    const float* query    = (const float*)d_in[0];   // [1,1024] f32
    const float* features = (const float*)d_in[1];   // [200000,1024] f32

    const int n    = in_sizes[1] / DIM;              // 200000
    const int topk = out_size / 2;                   // 100 values + 100 ids

    float* sims = (float*)d_ws;                      // n floats scratch (800 KB)
    float* outV = (float*)d_out;                     // values first ...
    int*   outI = (int*)d_out + topk;                // ... then int32 ids

    sim_kernel<<<(n + 127) / 128, 256, 0, stream>>>(query, features, sims, n);
    topk_kernel<<<1, 1024, 0, stream>>>(sims, outV, outI, n, topk);
}